// PickupCPW_87746181857525
// MI455X (gfx1250) — compile-verified
//
#include <hip/hip_runtime.h>
#include <math.h>

typedef float v2f __attribute__((ext_vector_type(2)));
typedef float v8f __attribute__((ext_vector_type(8)));

#define NXg   256
#define NYg   256
#define NTg   10
#define NFIL  12
#define KTOT  65536              // NX*NY
#define KBLOCKS 64
#define KPB   (KTOT / KBLOCKS)   // 1024 K per workgroup
#define NWAVES 8
#define KPW   (KPB / NWAVES)     // 128 K per wave -> 32 wmma steps

#define DXc   20.0e-9f
#define Z2c   1.0e-14f           // (100e-9)^2
// MU0/(4pi) * DX*DY*DZ * (DY/DT) = 1e-7 * 8e-24 * 4000 = 3.2e-27
#define SCALEc 3.2e-27f

__constant__ int c_xpos[NFIL] = {40, 48, 56, 64, 112, 120, 128, 136, 184, 192, 200, 208};

// ---------------------------------------------------------------------------
// Kernel 1: dS[t][ij] = -(2*Msat/DX) * grad_x( m[t+1,2] - m[t,2] )  (t = 0..8)
// rows 9..15 zero-padded so the WMMA A-matrix needs no masking.
// ---------------------------------------------------------------------------
__global__ __launch_bounds__(256) void build_dS(const float* __restrict__ m,
                                                const float* __restrict__ msat_p,
                                                float* __restrict__ dS) {
    const int idx = blockIdx.x * 256 + threadIdx.x;   // 16 * 65536 threads
    const int t  = idx >> 16;
    const int ij = idx & 65535;
    float val = 0.0f;
    if (t < NTg - 1) {
        const int i = ij >> 8, j = ij & 255;
        const float msat = msat_p[0];
        const float* m0 = m + (((t     * 3) + 2) << 16);  // m[t  ,2,:,:]
        const float* m1 = m + ((((t+1) * 3) + 2) << 16);  // m[t+1,2,:,:]
        auto w = [&](int ii) { int o = (ii << 8) | j; return m1[o] - m0[o]; };
        float g;
        if (i == 0)            g = w(1) - w(0);                    // one-sided
        else if (i == NXg - 1) g = w(NXg - 1) - w(NXg - 2);
        else                   g = 0.5f * (w(i + 1) - w(i - 1));   // central
        val = (-2.0f * msat / DXc) * g;
    }
    dS[idx] = val;
}

// ---------------------------------------------------------------------------
// Kernel 2: G[ij][f] = sum_k 1/sqrt((x_i - xfil_f)^2 + ((j-k)*DY)^2 + z^2)
// stored K-major x N (row-major [65536][16]); cols 12..15 zero-padded.
// ---------------------------------------------------------------------------
__global__ __launch_bounds__(256) void build_G(float* __restrict__ G) {
    const int idx = blockIdx.x * 256 + threadIdx.x;   // 65536 * 16 threads
    const int f  = idx & 15;
    const int ij = idx >> 4;
    float s = 0.0f;
    if (f < NFIL) {
        const int i = ij >> 8, j = ij & 255;
        const float dx   = (float)(i - c_xpos[f]) * DXc;
        const float dx2z = dx * dx + Z2c;
        for (int k = 0; k < NYg; ++k) {
            const float dy = (float)(j - k) * DXc;
            s += 1.0f / sqrtf(dx2z + dy * dy);
        }
    }
    G[idx] = s;
}

// ---------------------------------------------------------------------------
// Kernel 3: WMMA GEMM partials.  D = A(16 x 65536) * B(65536 x 16), fp32.
// Each workgroup owns a K-slab of 1024; its 8 waves run 32
// v_wmma_f32_16x16x4_f32 steps as 8 trips x 4 INDEPENDENT accumulators so
// consecutive WMMAs never chain D->C (XDL RAW hazard, ISA 7.12.1),
// LDS-reduce across waves, write one 16x16 partial tile.
// A-frag (16x4 f32, lane L): M = L&15, K = v + 2*(L>>4) per VGPR v.
// B-frag mirrors with N = L&15.  C/D: VGPR r <-> M = r + 8*(L>>4), N = L&15.
// ---------------------------------------------------------------------------
__global__ __launch_bounds__(256) void gemm_partial(const float* __restrict__ dS,
                                                    const float* __restrict__ G,
                                                    float* __restrict__ partial) {
    __shared__ float red[NWAVES * 256];
    const int tid  = threadIdx.x;
    const int wave = tid >> 5;
    const int lane = tid & 31;
    const int row  = lane & 15;          // M for A-frag, N for B-frag
    const int koff = (lane >> 4) << 1;   // hi half-wave covers K+2, K+3
    const int kbase = blockIdx.x * KPB + wave * KPW;

    v8f acc[4];
#pragma unroll
    for (int u = 0; u < 4; ++u) acc[u] = (v8f){0.f,0.f,0.f,0.f,0.f,0.f,0.f,0.f};

    const float* arow = dS + row * KTOT;
    for (int k = kbase; k < kbase + KPW; k += 16) {
        if (k + 256 < KTOT) {                       // wave-uniform: EXEC stays ~0
            __builtin_prefetch(&G[(k + 256) * 16], 0, 1);   // global_prefetch_b8
        }
#pragma unroll
        for (int u = 0; u < 4; ++u) {
            const int kk = k + 4 * u;
            v2f a, b;
            a.x = arow[kk + koff];
            a.y = arow[kk + koff + 1];
            b.x = G[(kk + koff)     * 16 + row];
            b.y = G[(kk + koff + 1) * 16 + row];
            acc[u] = __builtin_amdgcn_wmma_f32_16x16x4_f32(
                         /*neg_a=*/false, a, /*neg_b=*/false, b,
                         /*c_mod=*/(short)0, acc[u],
                         /*reuse_a=*/false, /*reuse_b=*/false);
        }
    }

#pragma unroll
    for (int r = 0; r < 8; ++r) {
        const float cr = acc[0][r] + acc[1][r] + acc[2][r] + acc[3][r];
        red[wave * 256 + r * 32 + lane] = cr;
    }
    __syncthreads();
    if (wave == 0) {
#pragma unroll
        for (int r = 0; r < 8; ++r) {
            float s = 0.0f;
            for (int w = 0; w < NWAVES; ++w) s += red[w * 256 + r * 32 + lane];
            partial[blockIdx.x * 256 + r * 32 + lane] = s;   // fragment layout
        }
    }
}

// ---------------------------------------------------------------------------
// Kernel 4: fixed-order partial reduction -> deltaV[9][12], then the
// single-frequency sin/cos fit -> ffit[9][12].  out = [deltaV | ffit].
// ---------------------------------------------------------------------------
__global__ __launch_bounds__(256) void finalize(const float* __restrict__ partial,
                                                float* __restrict__ out) {
    __shared__ float full[16][16];       // full16[M=t][N=f] = deltaV (padded)
    const int tid  = threadIdx.x;
    const int r    = tid >> 5;
    const int lane = tid & 31;
    const int M    = r + ((lane >> 4) << 3);
    const int Ncol = lane & 15;
    float s = 0.0f;
    for (int b = 0; b < KBLOCKS; ++b) s += partial[b * 256 + r * 32 + lane];
    full[M][Ncol] = s * SCALEc;
    __syncthreads();

    if (tid < 108) {
        const int t = tid / 12, f = tid % 12;
        out[tid] = full[t][f];                       // deltaV

        const float twopif0 = 6.2831853071795865f * 6.0e9f;
        float re = 0.0f, im = 0.0f;
        for (int tt = 0; tt < 9; ++tt) {             // t = (2991..2999)*DT
            const float arg = twopif0 * ((float)(2991 + tt) * 5.0e-12f);
            re += cosf(arg) * full[tt][f];
            im += sinf(arg) * full[tt][f];
        }
        re *= (2.0f / 9.0f);                         // norm == 2/L
        im *= (2.0f / 9.0f);
        const float argt = twopif0 * ((float)(2991 + t) * 5.0e-12f);
        out[108 + tid] = cosf(argt) * re + sinf(argt) * im;   // ffit
    }
}

// ---------------------------------------------------------------------------
extern "C" void kernel_launch(void* const* d_in, const int* in_sizes, int n_in,
                              void* d_out, int out_size, void* d_ws, size_t ws_size,
                              hipStream_t stream) {
    const float* m    = (const float*)d_in[0];   // [10,3,256,256]
    const float* msat = (const float*)d_in[1];   // scalar
    float* out = (float*)d_out;                  // 108 deltaV + 108 ffit

    float* ws      = (float*)d_ws;
    float* dS      = ws;                          // 16 * 65536 floats (4 MB)
    float* G       = ws + 16 * KTOT;              // 65536 * 16 floats (4 MB)
    float* partial = ws + 32 * KTOT;              // 64 * 256 floats  (64 KB)

    build_dS   <<<4096, 256, 0, stream>>>(m, msat, dS);
    build_G    <<<4096, 256, 0, stream>>>(G);
    gemm_partial<<<KBLOCKS, 256, 0, stream>>>(dS, G, partial);
    finalize   <<<1, 256, 0, stream>>>(partial, out);
}